// Abbott_74440373174275
// MI455X (gfx1250) — compile-verified
//
#include <hip/hip_runtime.h>

typedef __attribute__((ext_vector_type(2))) float v2f;
typedef __attribute__((ext_vector_type(8))) float v8f;

// One wave computes, for fixed pre-neuron e and a 16-wide post-neuron tile o0..o0+15:
//   S_pot[16b x 16o] = xbar_pre^T[16b x 8d] * dmap[8d x 16o]   (2x WMMA f32 16x16x4)
//   S_dep[16b x 16o] = Xd^T     [16b x 8d] * dmap[8d x 16o]   (2x WMMA, B reused)
// then fuses the W read / out write / clamped W_new write.
__global__ __launch_bounds__(32) void plasticity_wmma_kernel(
    const float* __restrict__ Xd,        // (8,16,N)
    const float* __restrict__ Xpost,     // (16,N)
    const float* __restrict__ W,         // (16,N,N)
    const float* __restrict__ xbar_pre,  // (8,16,N)
    const float* __restrict__ xbar_post, // (16,N)
    const float* __restrict__ dmap,      // (8,N,N)
    const float* __restrict__ A_p,       // (N,N)
    const float* __restrict__ A_d,       // (N,N)
    float* __restrict__ out,             // (16,N,N)
    float* __restrict__ W_new,           // (16,N,N)
    int N)
{
    const int e    = blockIdx.y;
    const int o0   = blockIdx.x << 4;
    const int lane = threadIdx.x;   // 0..31 (wave32)
    const int h    = lane >> 4;     // half-wave select
    const int m    = lane & 15;     // A: M row (batch) / B,C,D: N col

    const int NN = N * N;
    const int bN = 16 * N;

    // K (delay) indices owned by this lane for the two K=4 WMMA steps.
    const int d0 = 2 * h;
    const int d1 = 2 * h + 1;
    const int d2 = 4 + 2 * h;
    const int d3 = 5 + 2 * h;

    // ---- A matrices: 16x4 f32 layout (lanes 0-15: K=0/1, lanes 16-31: K=2/3) ----
    const int aoff = m * N + e;                 // [d, b=m, e]
    v2f a_pot0, a_pot1, a_dep0, a_dep1;
    a_pot0.x = xbar_pre[d0 * bN + aoff];
    a_pot0.y = xbar_pre[d1 * bN + aoff];
    a_pot1.x = xbar_pre[d2 * bN + aoff];
    a_pot1.y = xbar_pre[d3 * bN + aoff];
    a_dep0.x = Xd[d0 * bN + aoff];
    a_dep0.y = Xd[d1 * bN + aoff];
    a_dep1.x = Xd[d2 * bN + aoff];
    a_dep1.y = Xd[d3 * bN + aoff];

    // ---- B matrix: 4x16 f32, shared between both products (dmap[d, e, o]) ----
    const int boff = e * N + o0 + m;            // col n = m, contiguous across lanes
    v2f b0, b1;
    b0.x = dmap[(size_t)d0 * NN + boff];
    b0.y = dmap[(size_t)d1 * NN + boff];
    b1.x = dmap[(size_t)d2 * NN + boff];
    b1.y = dmap[(size_t)d3 * NN + boff];

    // ---- WMMA accumulation: K=8 as two chained K=4 steps ----
    v8f s_pot = {};
    v8f s_dep = {};
    s_pot = __builtin_amdgcn_wmma_f32_16x16x4_f32(false, a_pot0, false, b0,
                                                  (short)0, s_pot, false, false);
    s_pot = __builtin_amdgcn_wmma_f32_16x16x4_f32(false, a_pot1, false, b1,
                                                  (short)0, s_pot, false, false);
    s_dep = __builtin_amdgcn_wmma_f32_16x16x4_f32(false, a_dep0, false, b0,
                                                  (short)0, s_dep, false, false);
    s_dep = __builtin_amdgcn_wmma_f32_16x16x4_f32(false, a_dep1, false, b1,
                                                  (short)0, s_dep, false, false);

    // ---- Fused epilogue over the 8 batch rows this lane owns (M = r + 8h) ----
    const int   o  = o0 + m;
    const float ap = A_p[e * N + o];
    const float ad = A_d[e * N + o];
#pragma unroll
    for (int r = 0; r < 8; ++r) {
        const int    bi  = r + 8 * h;
        const float  pot = s_pot[r] * Xpost[bi * N + o] * ap;
        const float  dep = s_dep[r] * xbar_post[bi * N + o] * ad;
        const size_t idx = (size_t)bi * NN + (size_t)e * N + o;
        const float  w   = W[idx];
        out[idx] = w;                          // out = W before update
        float wn = w + pot - dep;
        wn = fminf(fmaxf(wn, 0.0f), 1.0f);     // clip to [0, WMAX]
        W_new[idx] = wn;
    }
}

// xbar_new = alpha * xbar + (1 - alpha) * x
__global__ void trace_update_kernel(const float* __restrict__ x,
                                    const float* __restrict__ xbar,
                                    float* __restrict__ outp,
                                    float alpha, int n) {
    int i = blockIdx.x * blockDim.x + threadIdx.x;
    if (i < n) outp[i] = alpha * xbar[i] + (1.0f - alpha) * x[i];
}

extern "C" void kernel_launch(void* const* d_in, const int* in_sizes, int n_in,
                              void* d_out, int out_size, void* d_ws, size_t ws_size,
                              hipStream_t stream) {
    // setup_inputs() order: Xd, Xpost, W, xbar_pre, xbar_post, dmap, A_p, A_d
    const float* Xd        = (const float*)d_in[0];
    const float* Xpost     = (const float*)d_in[1];
    const float* W         = (const float*)d_in[2];
    const float* xbar_pre  = (const float*)d_in[3];
    const float* xbar_post = (const float*)d_in[4];
    const float* dmap      = (const float*)d_in[5];
    const float* A_p       = (const float*)d_in[6];
    const float* A_d       = (const float*)d_in[7];

    const int NN = in_sizes[6];                 // N*N (A_p)
    int N = 1;
    while ((long)N * N < (long)NN) ++N;         // N = 1024
    const int b = in_sizes[1] / N;              // 16

    // d_out = concat(out, W_new, xbar_pre_new, xbar_post_new)
    float* out      = (float*)d_out;
    float* W_new    = out + (size_t)b * NN;
    float* xbp_new  = W_new + (size_t)b * NN;
    float* xbo_new  = xbp_new + (size_t)in_sizes[3];

    dim3 grid(N / 16, N);                       // (o-tiles, e)
    plasticity_wmma_kernel<<<grid, 32, 0, stream>>>(
        Xd, Xpost, W, xbar_pre, xbar_post, dmap, A_p, A_d, out, W_new, N);

    const int n_pre = in_sizes[0];              // d*b*N
    trace_update_kernel<<<(n_pre + 255) / 256, 256, 0, stream>>>(
        Xd, xbar_pre, xbp_new, 0.95f, n_pre);

    const int n_post = in_sizes[1];             // b*N
    trace_update_kernel<<<(n_post + 255) / 256, 256, 0, stream>>>(
        Xpost, xbar_post, xbo_new, 0.9f, n_post);
}